// MultiScaleTANCA_84026740179758
// MI455X (gfx1250) — compile-verified
//
#include <hip/hip_runtime.h>
#include <math.h>

// Problem constants (match reference)
#define BB 32
#define NN 128
#define HH 8
#define EE 64
#define CC 16
#define DD (HH * EE)   // 512
#define D4 (DD / 4)    // 128

typedef __attribute__((ext_vector_type(2))) float v2f;
typedef __attribute__((ext_vector_type(4))) float v4f;
typedef __attribute__((ext_vector_type(8))) float v8f;

// -------------------------------------------------------------------------
// Kernel 1: per-(b,c) gate = sigmoid( gelu(mean_n(values) @ w1 + b1) @ w2 + b2 )
// grid = B*C blocks, 256 threads
// -------------------------------------------------------------------------
__global__ __launch_bounds__(256)
void tanca_gate_kernel(const float* __restrict__ values,
                       const float* __restrict__ w1,
                       const float* __restrict__ b1,
                       const float* __restrict__ w2,
                       const float* __restrict__ b2,
                       float* __restrict__ gate) {
  __shared__ float vpool[DD];
  __shared__ float red[256];

  const int bc = blockIdx.x;                 // b*C + c
  const float* vbase = values + (size_t)bc * NN * DD;

  // mean over N (stride-D accesses, coalesced across d)
  for (int d = threadIdx.x; d < DD; d += blockDim.x) {
    float acc = 0.0f;
    #pragma unroll 4
    for (int n = 0; n < NN; ++n) acc += vbase[(size_t)n * DD + d];
    vpool[d] = acc * (1.0f / (float)NN);
  }
  __syncthreads();

  // hidden layer + exact GELU + dot with w2
  float partial = 0.0f;
  for (int j = threadIdx.x; j < D4; j += blockDim.x) {
    float hsum = b1[j];
    #pragma unroll 8
    for (int d = 0; d < DD; ++d) hsum += vpool[d] * w1[d * D4 + j];
    float g = 0.5f * hsum * (1.0f + erff(hsum * 0.7071067811865475f));
    partial += g * w2[j];
  }
  red[threadIdx.x] = partial;
  __syncthreads();
  for (int s = blockDim.x >> 1; s > 0; s >>= 1) {
    if ((int)threadIdx.x < s) red[threadIdx.x] += red[threadIdx.x + s];
    __syncthreads();
  }
  if (threadIdx.x == 0) {
    float z = red[0] + b2[0];
    gate[bc] = 1.0f / (1.0f + expf(-z));
  }
}

// -------------------------------------------------------------------------
// Async memory->LDS copy of 16 bytes per lane (gfx1250 ASYNCcnt path).
// ISA 15.18.3 opcode 98 GLOBAL_LOAD_ASYNC_TO_LDS_B128, GV addressing mode:
//   vdst = LDS byte address (addr[31:0] of generic shared pointer)
//   vaddr = 64-bit global address
// -------------------------------------------------------------------------
__device__ __forceinline__ void async_copy16(const float* gsrc, float* ldst) {
  unsigned lds_addr = (unsigned)(size_t)(void*)ldst;      // low 32 bits give the LDS byte address
  unsigned long long gaddr = (unsigned long long)(size_t)gsrc;
  asm volatile("global_load_async_to_lds_b128 %0, %1, off"
               :: "v"(lds_addr), "v"(gaddr)
               : "memory");
}

__device__ __forceinline__ void wait_async0() {
  asm volatile("s_wait_asynccnt 0x0" ::: "memory");
}

// -------------------------------------------------------------------------
// Kernel 2: masked neighborhood attention with gated V, f32 WMMA.
// grid = B*H blocks, 256 threads (8 waves). Wave w owns query rows
// [16w, 16w+16). Key tiles of 16 are staged via async DMA into
// double-buffered LDS; gate is folded into the P tile (g * P) so V is a
// raw copy:   sum_c P_c (g_c V_c) == sum_c (g_c P_c) V_c .
// -------------------------------------------------------------------------
#define KSTR 68   // padded LDS row stride (floats) for K/V tiles: bank-safe
#define PSTR 18   // padded LDS row stride (floats) for P tiles: 8B-aligned rows
#define NTILES ((CC * NN) / 16)   // 128

__global__ __launch_bounds__(256)
void tanca_attn_kernel(const float* __restrict__ q,
                       const float* __restrict__ keys,
                       const float* __restrict__ values,
                       const unsigned char* __restrict__ na_mask,
                       const float* __restrict__ gate,
                       float* __restrict__ out) {
  __shared__ __align__(16) float k_lds[2][16 * KSTR];
  __shared__ __align__(16) float v_lds[2][16 * KSTR];
  __shared__ __align__(16) float p_lds[8 * 16 * PSTR];

  const int bh   = blockIdx.x;
  const int b    = bh / HH;
  const int h    = bh % HH;
  const int tid  = threadIdx.x;
  const int wave = tid >> 5;      // 0..7 -> query tile
  const int lane = tid & 31;
  const int lo   = lane & 15;     // column / row-within-tile index
  const int hi   = lane >> 4;     // 0 or 1

  // cooperative-copy indices (one 16B async copy per thread per tile)
  const int ldrow = tid >> 4;     // 0..15
  const int ldc4  = tid & 15;     // 0..15
  const int ldcol = ldc4 * 4;

  // ---- preload Q as 16 A-fragments (16x4 f32 layout, ISA 7.12.2) ----
  // lane L: M = L%16 ; VGPR0/1 hold K = 2*hi, 2*hi+1 of each 4-chunk
  const float* qb =
      q + (((size_t)(b * NN + wave * 16 + lo) * HH) + h) * EE;
  v2f aq[16];
  #pragma unroll
  for (int kk = 0; kk < 16; ++kk)
    aq[kk] = *(const v2f*)(qb + kk * 4 + 2 * hi);

  v8f acc[4];
  #pragma unroll
  for (int t = 0; t < 4; ++t) acc[t] = (v8f){0.f,0.f,0.f,0.f,0.f,0.f,0.f,0.f};
  float rsum[8];
  #pragma unroll
  for (int r = 0; r < 8; ++r) rsum[r] = 0.0f;

  const unsigned char* mrow = na_mask + (size_t)h * NN * NN;
  float* p_w = p_lds + wave * 16 * PSTR;
  const float scale = 0.125f;     // E^-0.5, E=64

  // ---- prologue: async-stage tile 0 into buffer 0 ----
  {
    const size_t kvoff =
        (((size_t)(b * CC + 0) * NN + 0 + ldrow) * HH + h) * EE + ldcol;
    async_copy16(keys + kvoff,   &k_lds[0][ldrow * KSTR + ldcol]);
    async_copy16(values + kvoff, &v_lds[0][ldrow * KSTR + ldcol]);
  }
  wait_async0();
  __syncthreads();

  for (int j = 0; j < NTILES; ++j) {
    const int cur = j & 1;
    const int c   = j >> 3;                       // exo var
    const int nn0 = (j & 7) * 16;                 // key offset within N

    // ---- issue async DMA for next tile into the other buffer ----
    if (j + 1 < NTILES) {
      const int jn = j + 1, cn = jn >> 3, nn0n = (jn & 7) * 16;
      const size_t noff =
          (((size_t)(b * CC + cn) * NN + nn0n + ldrow) * HH + h) * EE + ldcol;
      async_copy16(keys + noff,   &k_lds[cur ^ 1][ldrow * KSTR + ldcol]);
      async_copy16(values + noff, &v_lds[cur ^ 1][ldrow * KSTR + ldcol]);
    }
    if (j + 2 < NTILES) {                         // L2 prefetch two tiles ahead
      const int jp = j + 2, cp = jp >> 3, nn0p = (jp & 7) * 16;
      const size_t poff =
          (((size_t)(b * CC + cp) * NN + nn0p + ldrow) * HH + h) * EE + ldcol;
      __builtin_prefetch((const void*)(keys + poff), 0, 0);
      __builtin_prefetch((const void*)(values + poff), 0, 0);
    }

    const float g = gate[b * CC + c];
    const float* kl = &k_lds[cur][0];
    const float* vl = &v_lds[cur][0];

    // ---- S = Q * K^T  (16 chained f32 WMMAs over E) ----
    v8f s = (v8f){0.f,0.f,0.f,0.f,0.f,0.f,0.f,0.f};
    #pragma unroll
    for (int kk = 0; kk < 16; ++kk) {
      // B frag (4xE x 16keys): lane L -> key lo, e = kk*4 + 2*hi (+1)
      v2f bk = *(const v2f*)(kl + lo * KSTR + kk * 4 + 2 * hi);
      s = __builtin_amdgcn_wmma_f32_16x16x4_f32(
              false, aq[kk], false, bk, (short)0, s, false, false);
    }

    // ---- mask, exp, row-sum (ungated), spill gated P to LDS ----
    #pragma unroll
    for (int r = 0; r < 8; ++r) {
      const int qn = wave * 16 + r + 8 * hi;      // D layout: M = r + 8*hi
      const int kn = nn0 + lo;
      const float p =
          mrow[qn * NN + kn] ? __expf(s[r] * scale) : 0.0f;
      float t = p;                                 // reduce within 16-lane half
      t += __shfl_xor(t, 1);
      t += __shfl_xor(t, 2);
      t += __shfl_xor(t, 4);
      t += __shfl_xor(t, 8);
      rsum[r] += t;
      p_w[(r + 8 * hi) * PSTR + lo] = p * g;       // gate folded into P
    }

    // ---- O += (gP) * V  (4 key-chunks x 4 e-column tiles) ----
    #pragma unroll
    for (int kc = 0; kc < 4; ++kc) {
      v2f ap = *(const v2f*)(p_w + lo * PSTR + kc * 4 + 2 * hi);
      #pragma unroll
      for (int t = 0; t < 4; ++t) {
        v2f bv;
        bv.x = vl[(kc * 4 + 2 * hi)     * KSTR + t * 16 + lo];
        bv.y = vl[(kc * 4 + 2 * hi + 1) * KSTR + t * 16 + lo];
        acc[t] = __builtin_amdgcn_wmma_f32_16x16x4_f32(
                     false, ap, false, bv, (short)0, acc[t], false, false);
      }
    }

    // ---- my async copies landed; all waves done with current buffers ----
    wait_async0();
    __syncthreads();
  }

  // ---- epilogue: normalize and store [B,N,H,E] ----
  #pragma unroll
  for (int t = 0; t < 4; ++t) {
    #pragma unroll
    for (int r = 0; r < 8; ++r) {
      const int qn = wave * 16 + r + 8 * hi;
      const float o = acc[t][r] / rsum[r];
      out[(((size_t)(b * NN + qn) * HH) + h) * EE + t * 16 + lo] = o;
    }
  }
}

// -------------------------------------------------------------------------
extern "C" void kernel_launch(void* const* d_in, const int* in_sizes, int n_in,
                              void* d_out, int out_size, void* d_ws, size_t ws_size,
                              hipStream_t stream) {
  const float* queries = (const float*)d_in[0];
  const float* keys    = (const float*)d_in[1];
  const float* values  = (const float*)d_in[2];
  const float* w1      = (const float*)d_in[3];
  const float* b1      = (const float*)d_in[4];
  const float* w2      = (const float*)d_in[5];
  const float* b2      = (const float*)d_in[6];
  const unsigned char* na_mask = (const unsigned char*)d_in[7];

  float* gate = (float*)d_ws;                 // B*C floats
  float* outp = (float*)d_out;

  tanca_gate_kernel<<<BB * CC, 256, 0, stream>>>(values, w1, b1, w2, b2, gate);
  tanca_attn_kernel<<<BB * HH, 256, 0, stream>>>(queries, keys, values,
                                                 na_mask, gate, outp);
}